// CANN_42554535969528
// MI455X (gfx1250) — compile-verified
//
#include <hip/hip_runtime.h>

#define NN 1680          // R*C
#define STEPS 14         // PRCN - 1

typedef float v2f __attribute__((ext_vector_type(2)));
typedef float v4f __attribute__((ext_vector_type(4)));
typedef float v8f __attribute__((ext_vector_type(8)));

__device__ __forceinline__ v8f wmma_pair(v4f a, v4f b, v8f acc) {
    v2f a0 = {a.x, a.y};
    v2f b0 = {b.x, b.y};
    acc = __builtin_amdgcn_wmma_f32_16x16x4_f32(
              false, a0, false, b0, (short)0, acc, false, false);
    v2f a1 = {a.z, a.w};
    v2f b1 = {b.z, b.w};
    acc = __builtin_amdgcn_wmma_f32_16x16x4_f32(
              false, a1, false, b1, (short)0, acc, false, false);
    return acc;
}

// ---------------------------------------------------------------------------
// temp = J @ r   via V_WMMA_F32_16X16X4_F32
//
// One workgroup (256 threads = 8 wave32 waves) per 16-row tile of J.
// Waves stride the K dimension in 8-column chunks. Per chunk each lane loads
// a float4 of J (global b128) and a float4 of r (LDS b128, half-wave
// broadcast), feeding two WMMAs (ISA 7.12.2 f32 16x4 A layout:
// lanes 0-15 carry K pair {0,1}, lanes 16-31 carry K pair {2,3}).
// B is column-broadcast so every D column duplicates the matvec partial
// (15/16 compute waste is irrelevant at 0.5 FLOP/byte).
// K loop is unrolled x2 with independent accumulators: both chunk loads are
// issued before a single wait, doubling per-wave MLP.
// ---------------------------------------------------------------------------
__global__ __launch_bounds__(256)
void cann_matvec_wmma(const float* __restrict__ J,
                      const float* __restrict__ r,
                      float* __restrict__ temp) {
    __shared__ __attribute__((aligned(16))) float rl[NN];
    __shared__ float red[8 * 16];   // [wave][row-in-tile] partials

    const int tid = threadIdx.x;

    // Stage r into LDS.
    for (int i = tid; i < NN; i += 256) rl[i] = r[i];
    __syncthreads();

    const int wave    = tid >> 5;      // 0..7
    const int lane    = tid & 31;
    const int lane_lo = lane & 15;
    const int half    = lane >> 4;     // 0: K pair {0,1}, 1: K pair {2,3}
    const int rowBase = blockIdx.x * 16;
    const float* Jrow = J + (size_t)(rowBase + lane_lo) * NN + half * 4;
    const float* rlh  = rl + half * 4;

    v8f acc0 = {0.f, 0.f, 0.f, 0.f, 0.f, 0.f, 0.f, 0.f};
    v8f acc1 = {0.f, 0.f, 0.f, 0.f, 0.f, 0.f, 0.f, 0.f};

    // 210 chunks of 8 columns; wave w owns chunks w, w+8, ...  (26 or 27).
    int c = wave;
    for (; c + 8 < NN / 8; c += 16) {
        // Issue all four loads, then consume (single wait, 4 WMMAs).
        v4f aA = *(const v4f*)(Jrow + c * 8);
        v4f aB = *(const v4f*)(Jrow + c * 8 + 64);
        v4f bA = *(const v4f*)(rlh + c * 8);
        v4f bB = *(const v4f*)(rlh + c * 8 + 64);
        acc0 = wmma_pair(aA, bA, acc0);
        acc1 = wmma_pair(aB, bB, acc1);
    }
    if (c < NN / 8) {   // odd tail chunk (waves 0,1 have 27 chunks)
        v4f aA = *(const v4f*)(Jrow + c * 8);
        v4f bA = *(const v4f*)(rlh + c * 8);
        acc0 = wmma_pair(aA, bA, acc0);
    }
#pragma unroll
    for (int v = 0; v < 8; ++v) acc0[v] += acc1[v];

    // D layout (32-bit C/D 16x16): VGPR v, lanes 0-15 -> M=v ; lanes 16-31 -> M=v+8.
    // All N-columns duplicate, so lanes 0 / 16 carry the full 16-row result.
    if (lane_lo == 0) {
#pragma unroll
        for (int v = 0; v < 8; ++v)
            red[wave * 16 + half * 8 + v] = acc0[v];
    }
    __syncthreads();

    // Deterministic fixed-order cross-wave reduction.
    if (tid < 16) {
        float s = 0.f;
#pragma unroll
        for (int w = 0; w < 8; ++w) s += red[w * 16 + tid];
        temp[rowBase + tid] = s;
    }
}

// ---------------------------------------------------------------------------
// U = temp + Iext ; sq = (0.2U)^2 ; recSum = 0.005*sum(sq) ; r_new = sq/recSum
// Single workgroup, deterministic tree reduction (bit-identical across replays).
// ---------------------------------------------------------------------------
__global__ __launch_bounds__(256)
void cann_update(const float* __restrict__ temp,
                 const float* __restrict__ Iext,
                 float* __restrict__ rOut,
                 float* __restrict__ uOut,
                 float* __restrict__ sumOut,
                 int writeExtras) {
    __shared__ float sq[NN];
    __shared__ float partial[256];

    const int tid = threadIdx.x;
    float local = 0.f;

    for (int i = tid; i < NN; i += 256) {
        float U = temp[i] + Iext[i];
        float s = (0.2f * U) * (0.2f * U);
        sq[i] = s;
        local += s;
        if (writeExtras) uOut[i] = U;
    }
    partial[tid] = local;
    __syncthreads();

#pragma unroll
    for (int s = 128; s > 0; s >>= 1) {
        if (tid < s) partial[tid] += partial[tid + s];
        __syncthreads();
    }

    const float recSum = 0.005f * partial[0];
    const float inv    = 1.0f / recSum;   // sq * (0.04/recSum)/0.04 == sq/recSum

    for (int i = tid; i < NN; i += 256)
        rOut[i] = sq[i] * inv;

    if (writeExtras && tid == 0) sumOut[0] = recSum;
}

// ---------------------------------------------------------------------------
// d_in[0] = net_in (2N fp32): [0,N) = Iext, [N,2N) = r0
// d_in[1] = J (N*N fp32)
// d_out   = [U_last (N), recSum (1), r_last (N)]  -> 3361 fp32
// d_ws    : temp (N) | rA (N) | rB (N)
// ---------------------------------------------------------------------------
extern "C" void kernel_launch(void* const* d_in, const int* in_sizes, int n_in,
                              void* d_out, int out_size, void* d_ws, size_t ws_size,
                              hipStream_t stream) {
    (void)in_sizes; (void)n_in; (void)out_size; (void)ws_size;

    const float* net_in = (const float*)d_in[0];
    const float* J      = (const float*)d_in[1];
    float*       out    = (float*)d_out;
    float*       ws     = (float*)d_ws;

    const float* Iext = net_in;          // first N
    const float* r0   = net_in + NN;     // second N
    float* temp = ws;
    float* rA   = ws + NN;
    float* rB   = ws + 2 * NN;

    float* uOut   = out;            // U_last
    float* sumOut = out + NN;       // recSum
    float* rFinal = out + NN + 1;   // r_last

    for (int s = 0; s < STEPS; ++s) {
        const bool  last = (s == STEPS - 1);
        const float* r_in  = (s == 0) ? r0 : ((s & 1) ? rA : rB);
        float*       r_out = last ? rFinal : ((s & 1) ? rB : rA);

        cann_matvec_wmma<<<NN / 16, 256, 0, stream>>>(J, r_in, temp);
        cann_update<<<1, 256, 0, stream>>>(temp, Iext, r_out, uOut, sumOut,
                                           last ? 1 : 0);
    }
}